// TRF_43654047596981
// MI455X (gfx1250) — compile-verified
//
#include <hip/hip_runtime.h>
#include <math.h>

#define B_     512
#define D_IN_  64
#define D_TRF_ 5120
#define P_     32
#define E_     160
#define D_OUT_ 3000
#define PAD_   61
#define D_CL_  4998

typedef __attribute__((ext_vector_type(2))) float v2f;
typedef __attribute__((ext_vector_type(8))) float v8f;

#define WMMA_F32(acc, a, b) \
    acc = __builtin_amdgcn_wmma_f32_16x16x4_f32(false, a, false, b, (short)0, acc, false, false)

// ---------------- 1) cl = (x*std+mean) @ mat, zero-padded to 5120 ----------
__global__ void prep_kernel(const float* __restrict__ x, const float* __restrict__ stdv,
                            const float* __restrict__ meanv, const float* __restrict__ mat,
                            float* __restrict__ cl) {
    int b = blockIdx.x;
    __shared__ float u[D_IN_];
    if (threadIdx.x < D_IN_)
        u[threadIdx.x] = x[b * D_IN_ + threadIdx.x] * stdv[threadIdx.x] + meanv[threadIdx.x];
    __syncthreads();
    for (int col = threadIdx.x; col < D_TRF_; col += blockDim.x) {
        float acc = 0.f;
        if (col >= PAD_ && col < PAD_ + D_CL_) {
            int c = col - PAD_;
#pragma unroll 8
            for (int k = 0; k < D_IN_; ++k)
                acc += u[k] * mat[(size_t)k * D_CL_ + c];
        }
        cl[(size_t)b * D_TRF_ + col] = acc;
    }
}

// ---------------- 2) LayerNorm over 5120 per row ----------------------------
__global__ void ln_kernel(const float* __restrict__ cl, const float* __restrict__ g,
                          const float* __restrict__ bta, float* __restrict__ xn) {
    int b = blockIdx.x;
    const float* row = cl + (size_t)b * D_TRF_;
    float s = 0.f, s2 = 0.f;
    for (int i = threadIdx.x; i < D_TRF_; i += blockDim.x) {
        float v = row[i];
        s += v; s2 += v * v;
    }
    __shared__ float r1[256], r2[256];
    r1[threadIdx.x] = s; r2[threadIdx.x] = s2;
    __syncthreads();
    for (int o = 128; o > 0; o >>= 1) {
        if (threadIdx.x < (unsigned)o) {
            r1[threadIdx.x] += r1[threadIdx.x + o];
            r2[threadIdx.x] += r2[threadIdx.x + o];
        }
        __syncthreads();
    }
    float mu  = r1[0] * (1.f / D_TRF_);
    float var = r2[0] * (1.f / D_TRF_) - mu * mu;
    float inv = rsqrtf(var + 1e-5f);
    for (int i = threadIdx.x; i < D_TRF_; i += blockDim.x)
        xn[(size_t)b * D_TRF_ + i] = (row[i] - mu) * inv * g[i] + bta[i];
}

// ------- 3) C[M,N] = A[M,K] @ W[N,K]^T + bias  (WMMA, 32x32 C per wave) -----
__global__ void gemm_nt_bias(const float* __restrict__ A, const float* __restrict__ W,
                             const float* __restrict__ bias, float* __restrict__ C,
                             int M, int Nc, int K) {
    int lane = threadIdx.x & 31;
    int wave = threadIdx.x >> 5;
    int ntn = (Nc + 31) >> 5;               // 32-col N blocks
    int mp  = M >> 5;                       // 32-row M blocks
    int tile = blockIdx.x * (blockDim.x >> 5) + wave;
    if (tile >= mp * ntn) return;
    int m0 = (tile / ntn) << 5;
    int n0 = (tile % ntn) << 5;
    int half = lane >> 4, l = lane & 15;

    const float* pa0 = A + (size_t)(m0 + l) * K + (half << 1);
    const float* pa1 = pa0 + (size_t)16 * K;
    int nb0 = n0 + l;      if (nb0 > Nc - 1) nb0 = Nc - 1;   // clamp: masked at store
    int nb1 = n0 + 16 + l; if (nb1 > Nc - 1) nb1 = Nc - 1;
    const float* pb0 = W + (size_t)nb0 * K + (half << 1);
    const float* pb1 = W + (size_t)nb1 * K + (half << 1);

    v8f acc00 = {}, acc01 = {}, acc10 = {}, acc11 = {};
    for (int k0 = 0; k0 < K; k0 += 16) {
        __builtin_prefetch(pa0 + k0 + 256, 0, 0);
        __builtin_prefetch(pb0 + k0 + 256, 0, 0);
        __builtin_prefetch(pb1 + k0 + 256, 0, 0);
#pragma unroll
        for (int kk = k0; kk < k0 + 16; kk += 4) {
            v2f b0 = { pb0[kk], pb0[kk + 1] };
            v2f b1 = { pb1[kk], pb1[kk + 1] };
            v2f a0 = { pa0[kk], pa0[kk + 1] };
            v2f a1 = { pa1[kk], pa1[kk + 1] };
            WMMA_F32(acc00, a0, b0);
            WMMA_F32(acc01, a0, b1);
            WMMA_F32(acc10, a1, b0);
            WMMA_F32(acc11, a1, b1);
        }
    }
    int col0 = n0 + l, col1 = n0 + 16 + l;
    float bv0 = bias[col0 < Nc ? col0 : 0];
    float bv1 = bias[col1 < Nc ? col1 : 0];
#pragma unroll
    for (int r = 0; r < 8; ++r) {
        int row  = m0 + (half << 3) + r;
        int row2 = row + 16;
        if (col0 < Nc) {
            C[(size_t)row  * Nc + col0] = acc00[r] + bv0;
            C[(size_t)row2 * Nc + col0] = acc10[r] + bv0;
        }
        if (col1 < Nc) {
            C[(size_t)row  * Nc + col1] = acc01[r] + bv1;
            C[(size_t)row2 * Nc + col1] = acc11[r] + bv1;
        }
    }
}

// ---------------- 4) attention: dot/softmax(dim=1)/AV + residual ------------
__global__ void attn_kernel(const float* __restrict__ q, const float* __restrict__ k,
                            const float* __restrict__ v, const float* __restrict__ cl,
                            float* __restrict__ h) {
    int b = blockIdx.x;
    const float* Q = q + (size_t)b * D_TRF_;
    const float* K = k + (size_t)b * D_TRF_;
    const float* V = v + (size_t)b * D_TRF_;
    __shared__ float qs[D_TRF_];
    __shared__ float ks[D_TRF_];
    __shared__ float att[P_][P_ + 1];
    for (int i = threadIdx.x; i < D_TRF_; i += blockDim.x) { qs[i] = Q[i]; ks[i] = K[i]; }
    __syncthreads();
    const float invs = 0.07905694150420948f;  // 1/sqrt(160)
    for (int idx = threadIdx.x; idx < P_ * P_; idx += blockDim.x) {
        int p = idx >> 5, qq = idx & 31;
        float s = 0.f;
#pragma unroll 8
        for (int e = 0; e < E_; ++e) s += qs[p * E_ + e] * ks[qq * E_ + e];
        att[p][qq] = s * invs;
    }
    __syncthreads();
    // softmax over the p axis (axis=1 of [B,P,Q]) -> per column qq
    if (threadIdx.x < P_) {
        int qq = threadIdx.x;
        float m = -1e30f;
        for (int p = 0; p < P_; ++p) m = fmaxf(m, att[p][qq]);
        float s = 0.f;
        for (int p = 0; p < P_; ++p) s += __expf(att[p][qq] - m);
        float inv = 1.f / s;
        for (int p = 0; p < P_; ++p) att[p][qq] = __expf(att[p][qq] - m) * inv;
    }
    __syncthreads();
    for (int f = threadIdx.x; f < D_TRF_; f += blockDim.x) {
        int p = f / E_, e = f % E_;
        float s = 0.f;
#pragma unroll
        for (int qq = 0; qq < P_; ++qq) s += att[p][qq] * V[qq * E_ + e];
        h[(size_t)b * D_TRF_ + f] = s + cl[(size_t)b * D_TRF_ + f];
    }
}

// ------- 5) o = (gain*h+bias) @ bt_w + bt_b; Supact + residual (32x32) ------
__global__ void gemm_nn_supact(const float* __restrict__ h, const float* __restrict__ W,
                               const float* __restrict__ bias,
                               const float* __restrict__ gain, const float* __restrict__ bsh,
                               const float* __restrict__ gamma, const float* __restrict__ beta,
                               float* __restrict__ y) {
    const int M = B_ * P_, Nc = E_, K = E_;
    int lane = threadIdx.x & 31;
    int wave = threadIdx.x >> 5;
    int ntn = Nc >> 5;                      // 5
    int mp  = M >> 5;
    int tile = blockIdx.x * (blockDim.x >> 5) + wave;
    if (tile >= mp * ntn) return;
    int m0 = (tile / ntn) << 5;
    int n0 = (tile % ntn) << 5;
    int half = lane >> 4, l = lane & 15;
    float sc = gain[0], sh = bsh[0];

    const float* pa0 = h + (size_t)(m0 + l) * K + (half << 1);
    const float* pa1 = pa0 + (size_t)16 * K;
    const float* pb0 = W + (size_t)(half << 1) * Nc + (n0 + l);        // NN: W[K,N]
    const float* pb1 = pb0 + 16;

    v8f acc00 = {}, acc01 = {}, acc10 = {}, acc11 = {};
    for (int k0 = 0; k0 < K; k0 += 16) {
#pragma unroll
        for (int kk = k0; kk < k0 + 16; kk += 4) {
            v2f b0 = { pb0[(size_t)kk * Nc], pb0[(size_t)(kk + 1) * Nc] };
            v2f b1 = { pb1[(size_t)kk * Nc], pb1[(size_t)(kk + 1) * Nc] };
            v2f a0 = { pa0[kk] * sc + sh,    pa0[kk + 1] * sc + sh };
            v2f a1 = { pa1[kk] * sc + sh,    pa1[kk + 1] * sc + sh };
            WMMA_F32(acc00, a0, b0);
            WMMA_F32(acc01, a0, b1);
            WMMA_F32(acc10, a1, b0);
            WMMA_F32(acc11, a1, b1);
        }
    }
#pragma unroll
    for (int t = 0; t < 4; ++t) {
        int row0 = m0 + ((t >> 1) << 4) + (half << 3);
        int col  = n0 + ((t & 1) << 4) + l;
        float bvc = bias[col];
        const v8f& acc = (t == 0) ? acc00 : (t == 1) ? acc01 : (t == 2) ? acc10 : acc11;
#pragma unroll
        for (int r = 0; r < 8; ++r) {
            int row = row0 + r;
            int f   = (row & (P_ - 1)) * E_ + col;   // feature index in [0,5120)
            float o  = acc[r] + bvc;
            float gm = gamma[f];
            float sig = 1.f / (1.f + __expf(-beta[f] * o));
            y[(size_t)row * E_ + col] = (gm + sig * (1.f - gm)) * o + h[(size_t)row * E_ + col];
        }
    }
}

// ------- 6) z = (y @ lin_w^T + lin_b)*out_gain + out_bias  (32x32 C) --------
__global__ void gemm_nt_out(const float* __restrict__ A, const float* __restrict__ W,
                            const float* __restrict__ bias,
                            const float* __restrict__ og, const float* __restrict__ ob,
                            float* __restrict__ C) {
    const int M = B_, Nc = D_OUT_, K = D_TRF_;
    int lane = threadIdx.x & 31;
    int wave = threadIdx.x >> 5;
    int ntn = (Nc + 31) >> 5;   // 94
    int mp  = M >> 5;           // 16
    int tile = blockIdx.x * (blockDim.x >> 5) + wave;
    if (tile >= mp * ntn) return;
    int m0 = (tile / ntn) << 5;
    int n0 = (tile % ntn) << 5;
    int half = lane >> 4, l = lane & 15;

    const float* pa0 = A + (size_t)(m0 + l) * K + (half << 1);
    const float* pa1 = pa0 + (size_t)16 * K;
    int nb0 = n0 + l;      if (nb0 > Nc - 1) nb0 = Nc - 1;
    int nb1 = n0 + 16 + l; if (nb1 > Nc - 1) nb1 = Nc - 1;
    const float* pb0 = W + (size_t)nb0 * K + (half << 1);
    const float* pb1 = W + (size_t)nb1 * K + (half << 1);

    v8f acc00 = {}, acc01 = {}, acc10 = {}, acc11 = {};
    for (int k0 = 0; k0 < K; k0 += 16) {
        __builtin_prefetch(pa0 + k0 + 256, 0, 0);
        __builtin_prefetch(pb0 + k0 + 256, 0, 0);
        __builtin_prefetch(pb1 + k0 + 256, 0, 0);
#pragma unroll
        for (int kk = k0; kk < k0 + 16; kk += 4) {
            v2f b0 = { pb0[kk], pb0[kk + 1] };
            v2f b1 = { pb1[kk], pb1[kk + 1] };
            v2f a0 = { pa0[kk], pa0[kk + 1] };
            v2f a1 = { pa1[kk], pa1[kk + 1] };
            WMMA_F32(acc00, a0, b0);
            WMMA_F32(acc01, a0, b1);
            WMMA_F32(acc10, a1, b0);
            WMMA_F32(acc11, a1, b1);
        }
    }
    int col0 = n0 + l, col1 = n0 + 16 + l;
    float gv = og[0], ov = ob[0];
    float bv0 = bias[col0 < Nc ? col0 : 0];
    float bv1 = bias[col1 < Nc ? col1 : 0];
#pragma unroll
    for (int r = 0; r < 8; ++r) {
        int row  = m0 + (half << 3) + r;
        int row2 = row + 16;
        if (col0 < Nc) {
            C[(size_t)row  * Nc + col0] = (acc00[r] + bv0) * gv + ov;
            C[(size_t)row2 * Nc + col0] = (acc10[r] + bv0) * gv + ov;
        }
        if (col1 < Nc) {
            C[(size_t)row  * Nc + col1] = (acc01[r] + bv1) * gv + ov;
            C[(size_t)row2 * Nc + col1] = (acc11[r] + bv1) * gv + ov;
        }
    }
}

// ---------------- host launcher ---------------------------------------------
extern "C" void kernel_launch(void* const* d_in, const int* in_sizes, int n_in,
                              void* d_out, int out_size, void* d_ws, size_t ws_size,
                              hipStream_t stream) {
    (void)in_sizes; (void)n_in; (void)out_size; (void)ws_size;
    const float* x        = (const float*)d_in[0];
    const float* stdv     = (const float*)d_in[1];
    const float* meanv    = (const float*)d_in[2];
    const float* mat      = (const float*)d_in[3];
    const float* ln_g     = (const float*)d_in[4];
    const float* ln_b     = (const float*)d_in[5];
    const float* wq       = (const float*)d_in[6];
    const float* bq       = (const float*)d_in[7];
    const float* wk       = (const float*)d_in[8];
    const float* bk       = (const float*)d_in[9];
    const float* wv       = (const float*)d_in[10];
    const float* bv       = (const float*)d_in[11];
    const float* bt_w     = (const float*)d_in[12];
    const float* bt_b     = (const float*)d_in[13];
    const float* bt_gain  = (const float*)d_in[14];
    const float* bt_bias  = (const float*)d_in[15];
    const float* sup_g    = (const float*)d_in[16];
    const float* sup_b    = (const float*)d_in[17];
    const float* lin_w    = (const float*)d_in[18];
    const float* lin_b    = (const float*)d_in[19];
    const float* out_gain = (const float*)d_in[20];
    const float* out_bias = (const float*)d_in[21];
    float* out = (float*)d_out;

    const size_t S = (size_t)B_ * D_TRF_;   // 2,621,440 floats per slot
    float* ws = (float*)d_ws;
    float* cl = ws + 0 * S;
    float* xn = ws + 1 * S;
    float* qb = ws + 2 * S;
    float* kb = ws + 3 * S;
    float* vb = ws + 4 * S;
    float* hb = ws + 5 * S;
    float* yb = xn;                          // xn dead after QKV; reuse for y

    prep_kernel<<<B_, 256, 0, stream>>>(x, stdv, meanv, mat, cl);
    ln_kernel<<<B_, 256, 0, stream>>>(cl, ln_g, ln_b, xn);

    // QKV: waves = (16384/32) * (160/32) = 512*5 = 2560 -> 320 blocks of 8 waves
    int waves_qkv = (B_ * P_ / 32) * (E_ / 32);
    gemm_nt_bias<<<waves_qkv / 8, 256, 0, stream>>>(xn, wq, bq, qb, B_ * P_, E_, E_);
    gemm_nt_bias<<<waves_qkv / 8, 256, 0, stream>>>(xn, wk, bk, kb, B_ * P_, E_, E_);
    gemm_nt_bias<<<waves_qkv / 8, 256, 0, stream>>>(xn, wv, bv, vb, B_ * P_, E_, E_);

    attn_kernel<<<B_, 256, 0, stream>>>(qb, kb, vb, cl, hb);

    gemm_nn_supact<<<waves_qkv / 8, 256, 0, stream>>>(hb, bt_w, bt_b, bt_gain, bt_bias,
                                                      sup_g, sup_b, yb);

    // out: waves = (512/32) * ceil(3000/32) = 16*94 = 1504 -> 188 blocks
    int waves_out = (B_ / 32) * ((D_OUT_ + 31) / 32);
    gemm_nt_out<<<waves_out / 8, 256, 0, stream>>>(yb, lin_w, lin_b, out_gain, out_bias, out);
}